// CostVolume_7241314861165
// MI455X (gfx1250) — compile-verified
//
#include <hip/hip_runtime.h>
#include <math.h>

// Problem constants (fixed by the reference's setup_inputs)
#define BB   2
#define CC_  64
#define HH   96
#define WW   320
#define GG   8
#define CPG  8           // channels per group
#define MAXD 48
#define DD   49          // MAXDISP + 1
#define EPSF 1e-5f

// One workgroup per (b, g, h): 320 threads, one pixel column each.
// Bandwidth-bound kernel: y row-slice staged in LDS via CDNA5 async
// global->LDS copies; x kept in registers (49x reuse); NT stores for
// the streaming 96 MB output.
__global__ __launch_bounds__(WW) void costvol_kernel(
    const float* __restrict__ x,
    const float* __restrict__ y,
    float* __restrict__ out)
{
    // Interleaved layout [j][c]: lane j's 8 channels are contiguous (32 B)
    // -> two ds_load_b128 per disparity in the sweep.
    __shared__ float yn_s[WW * CPG];            // 10240 B

    const int j = threadIdx.x;                  // pixel column 0..319
    int bid = blockIdx.x;                       // = (b*GG + g)*HH + h
    const int h = bid % HH;  bid /= HH;
    const int g = bid % GG;  bid /= GG;
    const int b = bid;

    const unsigned chan_stride = HH * WW;       // elements between channels
    const unsigned base = (((unsigned)(b * CC_ + g * CPG) * HH) + h) * WW + (unsigned)j;

    // ---- Phase 1a: async-copy y row slice (8 ch x 320 px) into LDS -------
    // global_load_async_to_lds_b32 vdst(lds byte addr), vaddr(u32 off), saddr
    // Tracked with ASYNCcnt (CDNA5 async path).
#pragma unroll
    for (int c = 0; c < CPG; ++c) {
        unsigned goff = (base + (unsigned)c * chan_stride) * 4u;    // byte offset vs y
        // Flat shared pointers carry the LDS byte offset in their low 32 bits
        // (ISA 10.2).
        unsigned loff = (unsigned)(unsigned long long)&yn_s[j * CPG + c];
        asm volatile("global_load_async_to_lds_b32 %0, %1, %2"
                     :: "v"(loff), "v"(goff), "s"(y)
                     : "memory");
    }

    // ---- Phase 1b (overlapped): load + normalize x into registers --------
    float xr[CPG];
    float ss = 0.0f;
#pragma unroll
    for (int c = 0; c < CPG; ++c) {
        xr[c] = __builtin_nontemporal_load(&x[base + (unsigned)c * chan_stride]);
        ss += xr[c] * xr[c];
    }
    const float xinv = 1.0f / (sqrtf(ss) + EPSF);
    float absx = 0.0f;
#pragma unroll
    for (int c = 0; c < CPG; ++c) {
        xr[c] *= xinv;
        absx += fabsf(xr[c]);
    }

    // ---- Phase 2: wait for this lane's async loads, normalize y in place -
    // Each lane touches only its own 8 LDS words, so no barrier is needed
    // before the in-place normalize; only before the cross-lane sweep.
    asm volatile("s_wait_asynccnt 0" ::: "memory");
    {
        float yr[CPG];
        float ys = 0.0f;
#pragma unroll
        for (int c = 0; c < CPG; ++c) {
            yr[c] = yn_s[j * CPG + c];
            ys += yr[c] * yr[c];
        }
        const float yinv = 1.0f / (sqrtf(ys) + EPSF);
#pragma unroll
        for (int c = 0; c < CPG; ++c)
            yn_s[j * CPG + c] = yr[c] * yinv;
    }
    __syncthreads();

    // ---- Phase 3: disparity sweep, NT streaming stores -------------------
    const float4* yv = reinterpret_cast<const float4*>(yn_s);
    float* outp = out + (((((unsigned)(b * GG + g)) * DD) * HH + h) * WW + (unsigned)j);
    for (int k = 0; k <= MAXD; ++k) {
        float cost;
        const int jm = j - k;
        if (jm >= 0) {
            const float4 a0 = yv[jm * 2 + 0];   // ds_load_b128
            const float4 a1 = yv[jm * 2 + 1];   // ds_load_b128
            cost = fabsf(xr[0] - a0.x) + fabsf(xr[1] - a0.y)
                 + fabsf(xr[2] - a0.z) + fabsf(xr[3] - a0.w)
                 + fabsf(xr[4] - a1.x) + fabsf(xr[5] - a1.y)
                 + fabsf(xr[6] - a1.z) + fabsf(xr[7] - a1.w);
        } else {
            // left zero padding: |xn - 0| summed over channels
            cost = absx;
        }
        __builtin_nontemporal_store(cost, outp);
        outp += HH * WW;
    }
}

extern "C" void kernel_launch(void* const* d_in, const int* in_sizes, int n_in,
                              void* d_out, int out_size, void* d_ws, size_t ws_size,
                              hipStream_t stream) {
    (void)in_sizes; (void)n_in; (void)d_ws; (void)ws_size; (void)out_size;
    const float* x = (const float*)d_in[0];
    const float* y = (const float*)d_in[1];
    float* out = (float*)d_out;

    const dim3 grid(BB * GG * HH);   // 1536 workgroups, one per (b,g,h)
    const dim3 block(WW);            // 320 threads = 10 wave32s
    costvol_kernel<<<grid, block, 0, stream>>>(x, y, out);
}